// HeteroRGCN_58265526338344
// MI455X (gfx1250) — compile-verified
//
#include <hip/hip_runtime.h>

// ---------------------------------------------------------------------------
// HeteroRGCN layer for MI455X (gfx1250).
//   out_gene = 0.5*(mean_int + mean_tgt) + h_gene@W_sl_g^T + b_sl_g
//   out_drug = mean_trt + h_drug@W_sl_d^T + b_sl_d
// Pipeline:
//   1) WMMA fp32 GEMMs: t_e = h_src@W_e^T + b_e  (workspace),
//      self-loop GEMMs written straight into d_out.
//   2) integer in-degree counts per etype (deterministic int atomics)
//   3) fused scatter: d_out[dst] += (scale/cnt[dst]) * t_e[src]  (fp32 atomics)
// Intermediates (~114 MB) + output (61 MB) fit the 192 MB L2, so the random
// row gathers in step 3 are L2-resident; HBM traffic ~ h reads + out writes.
// ---------------------------------------------------------------------------

typedef __attribute__((ext_vector_type(2))) float v2f;
typedef __attribute__((ext_vector_type(8))) float v8f;

#define D_DIM   128
#define NGENE   100000
#define NDRUG   20000
#define EINT    1000000
#define ETGT    400000
#define ETRT    400000

// C[m][n] = sum_k A[m][k]*W[n][k] + bias[n]; one wave per 16-row strip,
// 8 column tiles of 16, fp32 WMMA 16x16x4 over K=128 (32 k-steps).
__global__ __launch_bounds__(256) void gemm_awt_bias(
    const float* __restrict__ A, const float* __restrict__ W,
    const float* __restrict__ bias, float* __restrict__ C, int nStrips)
{
    int wave = (blockIdx.x << 3) + (threadIdx.x >> 5);   // 8 waves / block
    if (wave >= nStrips) return;                          // wave-uniform exit
    int l15  = threadIdx.x & 15;
    int half = (threadIdx.x >> 4) & 1;
    int mBase = wave << 4;

    const float* aRow = A + (size_t)(mBase + l15) * D_DIM + 2 * half;

    v8f acc[8];
#pragma unroll
    for (int n = 0; n < 8; ++n) acc[n] = (v8f){0.f,0.f,0.f,0.f,0.f,0.f,0.f,0.f};

    for (int k = 0; k < D_DIM; k += 4) {
        // A 16x4 fragment: lane m holds A[m][k+2*half], A[m][k+2*half+1]
        v2f a = *(const v2f*)(aRow + k);
#pragma unroll
        for (int n = 0; n < 8; ++n) {
            // B = W^T tile: B[kk][nn] = W[nBase+nn][kk]; lane = column nn
            const float* wRow = W + (size_t)(n * 16 + l15) * D_DIM + 2 * half;
            v2f b = *(const v2f*)(wRow + k);
            acc[n] = __builtin_amdgcn_wmma_f32_16x16x4_f32(
                false, a, false, b, (short)0, acc[n], false, false);
        }
    }

#pragma unroll
    for (int n = 0; n < 8; ++n) {
        int col = n * 16 + l15;
        float bv = bias[col];
#pragma unroll
        for (int r = 0; r < 8; ++r) {
            int row = mBase + r + 8 * half;               // C/D VGPR layout
            C[(size_t)row * D_DIM + col] = acc[n][r] + bv;
        }
    }
}

__global__ __launch_bounds__(256) void zero_i32(int* __restrict__ p, int n)
{
    int i = blockIdx.x * 256 + threadIdx.x;
    if (i < n) p[i] = 0;
}

__global__ __launch_bounds__(256) void count_edges(
    const int* __restrict__ dst, int* __restrict__ cnt, int n)
{
    int i = blockIdx.x * 256 + threadIdx.x;
    if (i < n) atomicAdd(&cnt[dst[i]], 1);
}

// One wave per edge; each lane moves 4 contiguous floats (128 = 32*4).
__global__ __launch_bounds__(256) void scatter_edges(
    const int* __restrict__ src, const int* __restrict__ dst,
    const int* __restrict__ cnt, const float* __restrict__ t,
    float* __restrict__ out, float scale, int nEdges)
{
    int tid = blockIdx.x * 256 + threadIdx.x;
    int e = tid >> 5;
    if (e >= nEdges) return;
    int l = tid & 31;

    int s = src[e];
    int d = dst[e];
    float w = scale / (float)cnt[d];    // cnt[d] >= 1 for any referenced dst

    const float4 v = *(const float4*)(t + (size_t)s * D_DIM + l * 4);
    float* o = out + (size_t)d * D_DIM + l * 4;
    atomicAdd(o + 0, w * v.x);
    atomicAdd(o + 1, w * v.y);
    atomicAdd(o + 2, w * v.z);
    atomicAdd(o + 3, w * v.w);
}

extern "C" void kernel_launch(void* const* d_in, const int* in_sizes, int n_in,
                              void* d_out, int out_size, void* d_ws, size_t ws_size,
                              hipStream_t stream)
{
    const float* h_gene    = (const float*)d_in[0];
    const float* h_drug    = (const float*)d_in[1];
    const float* W_int     = (const float*)d_in[2];
    const float* b_int     = (const float*)d_in[3];
    const float* W_tgt     = (const float*)d_in[4];
    const float* b_tgt     = (const float*)d_in[5];
    const float* W_trt     = (const float*)d_in[6];
    const float* b_trt     = (const float*)d_in[7];
    const float* W_sl_g    = (const float*)d_in[8];
    const float* b_sl_g    = (const float*)d_in[9];
    const float* W_sl_d    = (const float*)d_in[10];
    const float* b_sl_d    = (const float*)d_in[11];
    const int*   src_int   = (const int*)d_in[12];
    const int*   dst_int   = (const int*)d_in[13];
    const int*   src_tgt   = (const int*)d_in[14];
    const int*   dst_tgt   = (const int*)d_in[15];
    const int*   src_trt   = (const int*)d_in[16];
    const int*   dst_trt   = (const int*)d_in[17];

    // workspace layout (floats); offsets keep 16B alignment for float4 loads
    float* ws      = (float*)d_ws;
    float* t_int   = ws;                                   // 100000*128
    float* t_tgt   = ws + (size_t)NGENE * D_DIM;           //  20000*128
    float* t_trt   = t_tgt + (size_t)NDRUG * D_DIM;        // 100000*128
    int*   cnt_int = (int*)(t_trt + (size_t)NGENE * D_DIM);// 100000
    int*   cnt_tgt = cnt_int + NGENE;                      // 100000
    int*   cnt_trt = cnt_tgt + NGENE;                      //  20000

    float* out_gene = (float*)d_out;
    float* out_drug = out_gene + (size_t)NGENE * D_DIM;

    const int sg = NGENE / 16;   // 6250 strips
    const int sd = NDRUG / 16;   // 1250 strips
    dim3 blk(256);
    dim3 gridG((sg + 7) / 8);
    dim3 gridD((sd + 7) / 8);

    // 1) GEMMs (self-loop results go straight into d_out)
    gemm_awt_bias<<<gridG, blk, 0, stream>>>(h_gene, W_int,  b_int,  t_int,    sg);
    gemm_awt_bias<<<gridD, blk, 0, stream>>>(h_drug, W_tgt,  b_tgt,  t_tgt,    sd);
    gemm_awt_bias<<<gridG, blk, 0, stream>>>(h_gene, W_trt,  b_trt,  t_trt,    sg);
    gemm_awt_bias<<<gridG, blk, 0, stream>>>(h_gene, W_sl_g, b_sl_g, out_gene, sg);
    gemm_awt_bias<<<gridD, blk, 0, stream>>>(h_drug, W_sl_d, b_sl_d, out_drug, sd);

    // 2) in-degree counts (counts are contiguous: zero all 220000 in one go)
    const int nCnt = NGENE + NGENE + NDRUG;
    zero_i32<<<(nCnt + 255) / 256, blk, 0, stream>>>(cnt_int, nCnt);
    count_edges<<<(EINT + 255) / 256, blk, 0, stream>>>(dst_int, cnt_int, EINT);
    count_edges<<<(ETGT + 255) / 256, blk, 0, stream>>>(dst_tgt, cnt_tgt, ETGT);
    count_edges<<<(ETRT + 255) / 256, blk, 0, stream>>>(dst_trt, cnt_trt, ETRT);

    // 3) fused weighted scatter into d_out (one wave per edge)
    scatter_edges<<<(unsigned)(((size_t)EINT * 32 + 255) / 256), blk, 0, stream>>>(
        src_int, dst_int, cnt_int, t_int, out_gene, 0.5f, EINT);
    scatter_edges<<<(unsigned)(((size_t)ETGT * 32 + 255) / 256), blk, 0, stream>>>(
        src_tgt, dst_tgt, cnt_tgt, t_tgt, out_gene, 0.5f, ETGT);
    scatter_edges<<<(unsigned)(((size_t)ETRT * 32 + 255) / 256), blk, 0, stream>>>(
        src_trt, dst_trt, cnt_trt, t_trt, out_drug, 1.0f, ETRT);
}